// RecEncoder_DIN_76871324664067
// MI455X (gfx1250) — compile-verified
//
#include <hip/hip_runtime.h>
#include <math.h>

typedef __attribute__((ext_vector_type(2))) float v2f;
typedef __attribute__((ext_vector_type(8))) float v8f;
typedef __attribute__((ext_vector_type(4))) unsigned int v4u;
typedef __attribute__((ext_vector_type(8))) int v8i;
typedef __attribute__((ext_vector_type(4))) int v4i;

#define DEPS 1e-8f

#if defined(__HIP_DEVICE_COMPILE__) && defined(__has_builtin)
#if __has_builtin(__builtin_amdgcn_tensor_load_to_lds) && __has_builtin(__builtin_amdgcn_s_wait_tensorcnt)
#define HAVE_TDM 1
#endif
#endif
#ifndef HAVE_TDM
#define HAVE_TDM 0
#endif

static __device__ __forceinline__ float sigmoidf_(float x) {
    return 1.0f / (1.0f + __expf(-x));
}

#if HAVE_TDM
// Issue a TDM 2D tile load: 16-column slice of a row-major [rows x stride] f32
// matrix into LDS (compact [rows][16]). Columns >= tensor_cols read as zero.
// D# packing per CDNA5 ISA 8.3/8.4. Wave-uniform; call from one wave.
// clang-23 form: (v4u g0, v8i g1, v4i g2, v4i g3, v8i extra, i32 cpol)
static __device__ __forceinline__
void tdm_load_slice16(void* lds_dst, const float* gsrc, int tensor_cols,
                      int rows, int stride_elems)
{
    unsigned lds_off = (unsigned)(size_t)lds_dst;            // LDS byte offset
    unsigned long long ga = (unsigned long long)(size_t)gsrc;
    v4u g0;
    g0.x = 1u;                                               // count=1, user D#
    g0.y = lds_off;                                          // lds_addr
    g0.z = (unsigned)(ga & 0xFFFFFFFFull);                   // global_addr lo
    g0.w = (unsigned)((ga >> 32) & 0x01FFFFFFull) | (2u << 30); // addr hi | type=2
    v8i g1;
    g1[0] = (int)(2u << 16);                                 // data_size=4B, wg_mask=0
    g1[1] = (int)(((unsigned)tensor_cols & 0xFFFFu) << 16);  // tensor_dim0[15:0]
    g1[2] = (int)(((unsigned)rows & 0xFFFFu) << 16);         // td0 hi=0 | tensor_dim1 lo
    g1[3] = (int)(16u << 16);                                // td1 hi=0 | tile_dim0=16
    g1[4] = rows;                                            // tile_dim1=rows, tile_dim2=0
    g1[5] = stride_elems;                                    // tensor_dim0_stride lo
    g1[6] = 0;                                               // stride0 hi | stride1 lo
    g1[7] = 0;
    v4i z4 = {};
    v8i z8 = {};
    __builtin_amdgcn_tensor_load_to_lds(g0, g1, z4, z4, z8, 0);
}
#endif

// ---------------------------------------------------------------------------
// Kernel 0: zero stats accumulators
// ---------------------------------------------------------------------------
__global__ void k_zero(float* __restrict__ p, int n) {
    for (int i = threadIdx.x; i < n; i += blockDim.x) p[i] = 0.0f;
}

// ---------------------------------------------------------------------------
// Kernel 1: ai = [q, h, q-h, q*h] (B*L,256) @ W1(256,64) + b1 -> X1 (B*L,64)
// WMMA f32 16x16x4; A (q|h) staged per-wave in LDS; W1 16-col slice staged
// per n-tile via TDM; derived features synthesized in branch-free phases.
// ---------------------------------------------------------------------------
__global__ __launch_bounds__(128)
void k_gemm1(const int* __restrict__ sparse, const int* __restrict__ seqs,
             const float* __restrict__ item_table, const float* __restrict__ his_table,
             const float* __restrict__ W1, const float* __restrict__ b1,
             float* __restrict__ X1)
{
    __shared__ float at[4][16][132];     // 16 rows x (q[64]|h[64]), padded stride
    __shared__ float Ws[256 * 16];       // W1 column slice, [k][16]
    const int lane = threadIdx.x & 31;
    const int wave = threadIdx.x >> 5;
    const int rowBase = (blockIdx.x * 4 + wave) * 16;

    // Stage 16 rows x 128 floats (512 float4) per wave
    for (int j = lane; j < 512; j += 32) {
        int r   = j >> 5;
        int q4  = j & 31;                // <16: query part, else his part
        int row = rowBase + r;
        int b = row / 10, l = row - b * 10;
        const float4* src;
        int off4 = q4 & 15;
        if (q4 < 16) src = (const float4*)(item_table + (size_t)sparse[2 * b + 1] * 64);
        else         src = (const float4*)(his_table  + (size_t)seqs[b * 10 + l] * 64);
        float4 v = src[off4];
        float* dst = &at[wave][r][(q4 < 16 ? 0 : 64) + off4 * 4];
        dst[0] = v.x; dst[1] = v.y; dst[2] = v.z; dst[3] = v.w;
    }

    const int m    = lane & 15;
    const int half = lane >> 4;
    const int ka   = 2 * half;
    const float* Arow = at[wave][m];

    for (int nt = 0; nt < 4; ++nt) {
        const int n0 = nt * 16;
        __syncthreads();                 // prior compute done / A-stage done
#if HAVE_TDM
        if (wave == 0) {
            tdm_load_slice16(Ws, W1 + n0, 16, 256, 64);
            __builtin_amdgcn_s_wait_tensorcnt(0);
        }
#else
        for (int i = threadIdx.x; i < 256 * 16; i += 128) {
            int k = i >> 4, c = i & 15;
            Ws[i] = W1[(size_t)k * 64 + n0 + c];
        }
#endif
        __syncthreads();

        v8f acc = {};
        // phase 1: q (0..63) and h (64..127) stored directly
        #pragma unroll 4
        for (int k0 = 0; k0 < 128; k0 += 4) {
            int kk = k0 + ka;
            v2f a, bf;
            a.x = Arow[kk];     a.y = Arow[kk + 1];
            bf.x = Ws[kk * 16 + m]; bf.y = Ws[(kk + 1) * 16 + m];
            acc = __builtin_amdgcn_wmma_f32_16x16x4_f32(false, a, false, bf,
                                                        (short)0, acc, false, false);
        }
        // phase 2: q - h
        #pragma unroll 4
        for (int k0 = 128; k0 < 192; k0 += 4) {
            int kk = k0 + ka;
            v2f a, bf;
            a.x = Arow[kk - 128] - Arow[kk - 64];
            a.y = Arow[kk - 127] - Arow[kk - 63];
            bf.x = Ws[kk * 16 + m]; bf.y = Ws[(kk + 1) * 16 + m];
            acc = __builtin_amdgcn_wmma_f32_16x16x4_f32(false, a, false, bf,
                                                        (short)0, acc, false, false);
        }
        // phase 3: q * h
        #pragma unroll 4
        for (int k0 = 192; k0 < 256; k0 += 4) {
            int kk = k0 + ka;
            v2f a, bf;
            a.x = Arow[kk - 192] * Arow[kk - 128];
            a.y = Arow[kk - 191] * Arow[kk - 127];
            bf.x = Ws[kk * 16 + m]; bf.y = Ws[(kk + 1) * 16 + m];
            acc = __builtin_amdgcn_wmma_f32_16x16x4_f32(false, a, false, bf,
                                                        (short)0, acc, false, false);
        }
        float bias = b1[n0 + m];
        for (int r = 0; r < 8; ++r) {
            int mm = r + 8 * half;
            X1[(size_t)(rowBase + mm) * 64 + n0 + m] = acc[r] + bias;
        }
    }
}

// ---------------------------------------------------------------------------
// Generic per-channel sum / sum-of-squares reduction (cols <= 256)
// ---------------------------------------------------------------------------
__global__ __launch_bounds__(256)
void k_reduce(const float* __restrict__ X, int rows, int cols,
              float* __restrict__ sum, float* __restrict__ ssum)
{
    __shared__ float ls[256], lq[256];
    for (int c = threadIdx.x; c < cols; c += blockDim.x) { ls[c] = 0.f; lq[c] = 0.f; }
    __syncthreads();
    size_t total  = (size_t)rows * (size_t)cols;
    size_t stride = (size_t)gridDim.x * blockDim.x;
    for (size_t e = (size_t)blockIdx.x * blockDim.x + threadIdx.x; e < total; e += stride) {
        float v = X[e];
        int c = (int)(e % (size_t)cols);
        atomicAdd(&ls[c], v);
        atomicAdd(&lq[c], v * v);
    }
    __syncthreads();
    for (int c = threadIdx.x; c < cols; c += blockDim.x) {
        atomicAdd(&sum[c],  ls[c]);
        atomicAdd(&ssum[c], lq[c]);
    }
}

// ---------------------------------------------------------------------------
// Kernel 3: dice3(X1) then collapsed linear tail -> score (masked)
// score = dice3(x) . (W2@Wf) + (b2.Wf + bf)
// ---------------------------------------------------------------------------
__global__ __launch_bounds__(256)
void k_score(const float* __restrict__ X1, const int* __restrict__ seqs,
             const float* __restrict__ s1, const float* __restrict__ ss1,
             const float* __restrict__ g1, const float* __restrict__ be1,
             const float* __restrict__ a1,
             const float* __restrict__ W2, const float* __restrict__ b2,
             const float* __restrict__ Wf, const float* __restrict__ bf,
             float* __restrict__ SCORE, float invN)
{
    __shared__ float Weff[64], scp[64], shp[64], alp[64];
    __shared__ float beff;
    int tid = threadIdx.x;
    if (tid < 64) {
        float s = 0.f;
        for (int j = 0; j < 16; ++j) s += W2[tid * 16 + j] * Wf[j];
        Weff[tid] = s;
        float mu  = s1[tid] * invN;
        float var = ss1[tid] * invN - mu * mu;
        float rs  = rsqrtf(var + DEPS) * g1[tid];
        scp[tid] = rs; shp[tid] = be1[tid] - mu * rs; alp[tid] = a1[tid];
    }
    if (tid == 0) {
        float s = bf[0];
        for (int j = 0; j < 16; ++j) s += b2[j] * Wf[j];
        beff = s;
    }
    __syncthreads();
    int row = blockIdx.x * 256 + tid;
    float acc = beff;
    const float4* xr = (const float4*)(X1 + (size_t)row * 64);
    for (int c4 = 0; c4 < 16; ++c4) {
        float4 v = xr[c4];
        #pragma unroll
        for (int q = 0; q < 4; ++q) {
            int c = c4 * 4 + q;
            float x = (&v.x)[q];
            float p = sigmoidf_(x * scp[c] + shp[c]);
            acc += x * (p + alp[c] * (1.f - p)) * Weff[c];
        }
    }
    SCORE[row] = (seqs[row] == 0) ? 0.f : acc;
}

// ---------------------------------------------------------------------------
// Kernel 4: pooled[b,e] = sum_l score[b,l] * his[b,l,e]
// ---------------------------------------------------------------------------
__global__ __launch_bounds__(256)
void k_pool(const float* __restrict__ SCORE, const int* __restrict__ seqs,
            const float* __restrict__ his_table, float* __restrict__ pooled)
{
    int t = blockIdx.x * 256 + threadIdx.x;
    int b = t >> 6, e = t & 63;
    float acc = 0.f;
    for (int l = 0; l < 10; ++l) {
        float s = SCORE[b * 10 + l];
        acc += s * his_table[(size_t)seqs[b * 10 + l] * 64 + e];
    }
    pooled[(size_t)b * 64 + e] = acc;
}

// ---------------------------------------------------------------------------
// Kernel 5: cf = [item|pooled|user] (B,192) @ Wa(192,200) + ba -> ZA (B,200)
// Wa 16-col slices staged via TDM; OOB columns (200..207) read as zero.
// ---------------------------------------------------------------------------
__global__ __launch_bounds__(128)
void k_gemmA(const int* __restrict__ sparse,
             const float* __restrict__ user_table, const float* __restrict__ item_table,
             const float* __restrict__ pooled,
             const float* __restrict__ Wa, const float* __restrict__ ba,
             float* __restrict__ ZA)
{
    __shared__ float at[4][16][196];
    __shared__ float Ws[192 * 16];
    const int lane = threadIdx.x & 31;
    const int wave = threadIdx.x >> 5;
    const int rowBase = (blockIdx.x * 4 + wave) * 16;

    for (int j = lane; j < 768; j += 32) {      // 16 rows x 48 float4
        int r = j / 48, seg = j - r * 48;
        int row = rowBase + r;
        const float4* src;
        if (seg < 16)      src = (const float4*)(item_table + (size_t)sparse[2 * row + 1] * 64);
        else if (seg < 32) src = (const float4*)(pooled + (size_t)row * 64);
        else               src = (const float4*)(user_table + (size_t)sparse[2 * row] * 64);
        float4 v = src[seg & 15];
        float* dst = &at[wave][r][seg * 4];
        dst[0] = v.x; dst[1] = v.y; dst[2] = v.z; dst[3] = v.w;
    }

    const int m = lane & 15, half = lane >> 4;
    const int ka = 2 * half;
    const float* Arow = at[wave][m];

    for (int nt = 0; nt < 13; ++nt) {           // N = 200 padded to 208
        const int n0  = nt * 16;
        const int col = n0 + m;
        __syncthreads();
#if HAVE_TDM
        if (wave == 0) {
            tdm_load_slice16(Ws, Wa + n0, 200 - n0, 192, 200);
            __builtin_amdgcn_s_wait_tensorcnt(0);
        }
#else
        for (int i = threadIdx.x; i < 192 * 16; i += 128) {
            int k = i >> 4, c = i & 15;
            Ws[i] = (n0 + c < 200) ? Wa[(size_t)k * 200 + n0 + c] : 0.f;
        }
#endif
        __syncthreads();

        v8f acc = {};
        #pragma unroll 4
        for (int k0 = 0; k0 < 192; k0 += 4) {
            int kk = k0 + ka;
            v2f a, bf;
            a.x = Arow[kk]; a.y = Arow[kk + 1];
            bf.x = Ws[kk * 16 + m]; bf.y = Ws[(kk + 1) * 16 + m];
            acc = __builtin_amdgcn_wmma_f32_16x16x4_f32(false, a, false, bf,
                                                        (short)0, acc, false, false);
        }
        if (col < 200) {
            float bias = ba[col];
            for (int r = 0; r < 8; ++r)
                ZA[(size_t)(rowBase + r + 8 * half) * 200 + col] = acc[r] + bias;
        }
    }
}

// ---------------------------------------------------------------------------
// Kernel 7: z = dice2(ZA) (fused into A staging), then z @ Wb(200,80) + bb
// ---------------------------------------------------------------------------
__global__ __launch_bounds__(128)
void k_gemmB(const float* __restrict__ ZA,
             const float* __restrict__ sA, const float* __restrict__ ssA,
             const float* __restrict__ ga, const float* __restrict__ bea,
             const float* __restrict__ aa,
             const float* __restrict__ Wb, const float* __restrict__ bb,
             float* __restrict__ ZB, float invN)
{
    __shared__ float at[4][16][204];
    __shared__ float scp[200], shp[200], alp[200];
    const int tid = threadIdx.x;
    for (int c = tid; c < 200; c += 128) {
        float mu  = sA[c] * invN;
        float var = ssA[c] * invN - mu * mu;
        float rs  = rsqrtf(var + DEPS) * ga[c];
        scp[c] = rs; shp[c] = bea[c] - mu * rs; alp[c] = aa[c];
    }
    __syncthreads();
    const int lane = tid & 31, wave = tid >> 5;
    const int rowBase = (blockIdx.x * 4 + wave) * 16;

    for (int j = lane; j < 800; j += 32) {      // 16 rows x 50 float4
        int r = j / 50, seg = j - r * 50;
        int row = rowBase + r;
        float4 v = ((const float4*)(ZA + (size_t)row * 200))[seg];
        int c0 = seg * 4;
        float* dst = &at[wave][r][c0];
        #pragma unroll
        for (int q = 0; q < 4; ++q) {
            float x = (&v.x)[q];
            int c = c0 + q;
            float p = sigmoidf_(x * scp[c] + shp[c]);
            dst[q] = x * (p + alp[c] * (1.f - p));
        }
    }
    __syncthreads();

    const int m = lane & 15, half = lane >> 4;
    const int ka = 2 * half;
    const float* Arow = at[wave][m];
    for (int nt = 0; nt < 5; ++nt) {            // N = 80
        const int col = nt * 16 + m;
        v8f acc = {};
        #pragma unroll 4
        for (int k0 = 0; k0 < 200; k0 += 4) {
            int kk = k0 + ka;
            v2f a, bf;
            a.x = Arow[kk]; a.y = Arow[kk + 1];
            bf.x = Wb[(size_t)kk * 80 + col];
            bf.y = Wb[(size_t)(kk + 1) * 80 + col];
            acc = __builtin_amdgcn_wmma_f32_16x16x4_f32(false, a, false, bf,
                                                        (short)0, acc, false, false);
        }
        float bias = bb[col];
        for (int r = 0; r < 8; ++r)
            ZB[(size_t)(rowBase + r + 8 * half) * 80 + col] = acc[r] + bias;
    }
}

// ---------------------------------------------------------------------------
// Kernel 9: out = sigmoid(dice2(ZB) . Wc + bc)
// ---------------------------------------------------------------------------
__global__ __launch_bounds__(256)
void k_final(const float* __restrict__ ZB,
             const float* __restrict__ sB, const float* __restrict__ ssB,
             const float* __restrict__ gb, const float* __restrict__ beb,
             const float* __restrict__ ab,
             const float* __restrict__ Wc, const float* __restrict__ bc,
             float* __restrict__ out, float invN)
{
    __shared__ float scp[80], shp[80], alp[80], wcp[80];
    int tid = threadIdx.x;
    if (tid < 80) {
        float mu  = sB[tid] * invN;
        float var = ssB[tid] * invN - mu * mu;
        float rs  = rsqrtf(var + DEPS) * gb[tid];
        scp[tid] = rs; shp[tid] = beb[tid] - mu * rs; alp[tid] = ab[tid]; wcp[tid] = Wc[tid];
    }
    __syncthreads();
    int b = blockIdx.x * 256 + tid;
    float acc = bc[0];
    const float4* zr = (const float4*)(ZB + (size_t)b * 80);
    for (int c4 = 0; c4 < 20; ++c4) {
        float4 v = zr[c4];
        #pragma unroll
        for (int q = 0; q < 4; ++q) {
            int c = c4 * 4 + q;
            float x = (&v.x)[q];
            float p = sigmoidf_(x * scp[c] + shp[c]);
            acc += x * (p + alp[c] * (1.f - p)) * wcp[c];
        }
    }
    out[b] = sigmoidf_(acc);
}

// ---------------------------------------------------------------------------
extern "C" void kernel_launch(void* const* d_in, const int* in_sizes, int n_in,
                              void* d_out, int out_size, void* d_ws, size_t ws_size,
                              hipStream_t stream)
{
    const int*   sparse = (const int*)  d_in[0];
    const int*   seqs   = (const int*)  d_in[1];
    const float* user_t = (const float*)d_in[2];
    const float* item_t = (const float*)d_in[3];
    const float* his_t  = (const float*)d_in[4];
    const float* W1  = (const float*)d_in[5];
    const float* b1  = (const float*)d_in[6];
    const float* g1  = (const float*)d_in[7];
    const float* be1 = (const float*)d_in[8];
    const float* a1  = (const float*)d_in[9];
    const float* W2  = (const float*)d_in[10];
    const float* b2  = (const float*)d_in[11];
    const float* Wf  = (const float*)d_in[12];
    const float* bf  = (const float*)d_in[13];
    const float* Wa  = (const float*)d_in[14];
    const float* ba  = (const float*)d_in[15];
    const float* ga  = (const float*)d_in[16];
    const float* bea = (const float*)d_in[17];
    const float* aa  = (const float*)d_in[18];
    const float* Wb  = (const float*)d_in[19];
    const float* bb  = (const float*)d_in[20];
    const float* gb  = (const float*)d_in[21];
    const float* beb = (const float*)d_in[22];
    const float* ab  = (const float*)d_in[23];
    const float* Wc  = (const float*)d_in[24];
    const float* bc  = (const float*)d_in[25];
    float* out = (float*)d_out;

    const int B = 16384, L = 10;

    // workspace layout (floats)
    float* ws  = (float*)d_ws;
    float* X1  = ws;                          // B*L*64
    float* SC  = X1  + (size_t)B * L * 64;    // B*L
    float* PL  = SC  + (size_t)B * L;         // B*64
    float* ZA  = PL  + (size_t)B * 64;        // B*200
    float* ZB  = ZA  + (size_t)B * 200;       // B*80
    float* S1  = ZB  + (size_t)B * 80;        // stats: 64+64+200+200+80+80 = 688
    float* SS1 = S1  + 64;
    float* SA  = SS1 + 64;
    float* SSA = SA  + 200;
    float* SB  = SSA + 200;
    float* SSB = SB  + 80;

    k_zero<<<1, 256, 0, stream>>>(S1, 688);

    k_gemm1 <<<(B * L) / 64, 128, 0, stream>>>(sparse, seqs, item_t, his_t, W1, b1, X1);
    k_reduce<<<512, 256, 0, stream>>>(X1, B * L, 64, S1, SS1);
    k_score <<<(B * L) / 256, 256, 0, stream>>>(X1, seqs, S1, SS1, g1, be1, a1,
                                                W2, b2, Wf, bf, SC, 1.0f / (float)(B * L));
    k_pool  <<<(B * 64) / 256, 256, 0, stream>>>(SC, seqs, his_t, PL);

    k_gemmA <<<B / 64, 128, 0, stream>>>(sparse, user_t, item_t, PL, Wa, ba, ZA);
    k_reduce<<<256, 256, 0, stream>>>(ZA, B, 200, SA, SSA);
    k_gemmB <<<B / 64, 128, 0, stream>>>(ZA, SA, SSA, ga, bea, aa, Wb, bb, ZB, 1.0f / (float)B);
    k_reduce<<<256, 256, 0, stream>>>(ZB, B, 80, SB, SSB);
    k_final <<<B / 256, 256, 0, stream>>>(ZB, SB, SSB, gb, beb, ab, Wc, bc, out, 1.0f / (float)B);

    (void)in_sizes; (void)n_in; (void)out_size; (void)ws_size;
}